// GCNLayer_69784628625697
// MI455X (gfx1250) — compile-verified
//
#include <hip/hip_runtime.h>

typedef float v2f __attribute__((ext_vector_type(2)));
typedef float v8f __attribute__((ext_vector_type(8)));

#define DFEAT 128
#define LPAD  132   // padded LDS row stride (floats): 132 % 64 = 4 -> conflict-free column walks

// ---------------- Kernel 1: zero the aggregation workspace ----------------
__global__ __launch_bounds__(256) void gcn_zero_kernel(float4* __restrict__ p, long n4) {
    long i = (long)blockIdx.x * blockDim.x + threadIdx.x;
    if (i < n4) p[i] = make_float4(0.f, 0.f, 0.f, 0.f);
}

// ---------------- Kernel 2: gather src rows, atomic scatter-add to dst ----
// One wave (32 lanes) per edge: lane l handles 4 contiguous floats -> a 512B
// coalesced gather per edge, 4 L2-resident f32 atomics per lane.
__global__ __launch_bounds__(256) void gcn_scatter_kernel(
    const float* __restrict__ feat, const int* __restrict__ src,
    const int* __restrict__ dst, float* __restrict__ hagg, int E)
{
    int gwave = (int)(((long)blockIdx.x * blockDim.x + threadIdx.x) >> 5);
    int lane  = threadIdx.x & 31;
    if (gwave >= E) return;
    long s = src[gwave];
    long d = dst[gwave];
    const float4 v = *(const float4*)(feat + s * DFEAT + (lane << 2));
    float* p = hagg + d * DFEAT + (lane << 2);
    __hip_atomic_fetch_add(p + 0, v.x, __ATOMIC_RELAXED, __HIP_MEMORY_SCOPE_AGENT);
    __hip_atomic_fetch_add(p + 1, v.y, __ATOMIC_RELAXED, __HIP_MEMORY_SCOPE_AGENT);
    __hip_atomic_fetch_add(p + 2, v.z, __ATOMIC_RELAXED, __HIP_MEMORY_SCOPE_AGENT);
    __hip_atomic_fetch_add(p + 3, v.w, __ATOMIC_RELAXED, __HIP_MEMORY_SCOPE_AGENT);
}

// ---------------- Kernel 3: WMMA fp32 GEMM + snorm + LayerNorm + ReLU -----
// Block = 256 threads = 8 waves. Block computes rows [rowBase, rowBase+16).
// Wave w computes the 16x16 tile at columns [16w, 16w+16) via
// V_WMMA_F32_16X16X4_F32 over a K-loop of 32 steps.
__global__ __launch_bounds__(256) void gcn_gemm_norm_kernel(
    const float* __restrict__ hagg, const float* __restrict__ Wm,
    const float* __restrict__ snorm, const float* __restrict__ gscale,
    const float* __restrict__ gbias, float* __restrict__ out, int N)
{
    __shared__ float sT[16 * LPAD];   // H tile during GEMM, then output staging

    const int tid     = threadIdx.x;
    const int waveId  = tid >> 5;
    const int lane    = tid & 31;
    const int rowBase = blockIdx.x << 4;

    // Cooperatively stage the 16x128 H tile into padded LDS (coalesced float4).
    for (int i = tid; i < 512; i += 256) {
        int r  = i >> 5;          // tile row 0..15
        int c4 = i & 31;          // float4 column 0..31
        float4 v = make_float4(0.f, 0.f, 0.f, 0.f);
        if (rowBase + r < N)
            v = *(const float4*)(hagg + (long)(rowBase + r) * DFEAT + (c4 << 2));
        *(float4*)&sT[r * LPAD + (c4 << 2)] = v;
    }
    __syncthreads();

    const int half    = lane >> 4;   // 0: K lo pair, 1: K hi pair
    const int l16     = lane & 15;   // M for A-frag / N for B-frag
    const int colBase = waveId << 4;

    v8f c = {0.f, 0.f, 0.f, 0.f, 0.f, 0.f, 0.f, 0.f};
    for (int k = 0; k < DFEAT; k += 4) {
        const int kk = k + (half << 1);
        // A fragment: H[rowBase + l16][kk], H[...][kk+1]  (ISA 16x4 f32 layout)
        v2f a = *(const v2f*)&sT[l16 * LPAD + kk];
        // B fragment: (W^T)[kk][colBase+l16] = W[colBase+l16][kk]
        const float* wp = Wm + (long)(colBase + l16) * DFEAT + kk;
        v2f b;
        b.x = wp[0];
        b.y = wp[1];
        c = __builtin_amdgcn_wmma_f32_16x16x4_f32(
                /*neg_a=*/false, a, /*neg_b=*/false, b,
                /*c_mod=*/(short)0, c, /*reuse_a=*/false, /*reuse_b=*/false);
    }
    __syncthreads();   // all waves done reading sT -> safe to reuse

    // Stage D-matrix tiles (scaled by snorm) into LDS. C/D layout: VGPR v ->
    // row M = v + 8*half, col N = l16.
    for (int v = 0; v < 8; ++v) {
        int m   = v + (half << 3);
        int row = rowBase + m;
        float s = snorm[row < N ? row : (N - 1)];
        sT[m * LPAD + colBase + l16] = c[v] * s;
    }
    __syncthreads();

    // LayerNorm + ReLU: each wave owns two full 128-wide rows.
    for (int rr = 0; rr < 2; ++rr) {
        const int r = (waveId << 1) + rr;
        float x0 = sT[r * LPAD + lane];
        float x1 = sT[r * LPAD + lane + 32];
        float x2 = sT[r * LPAD + lane + 64];
        float x3 = sT[r * LPAD + lane + 96];
        float sum = x0 + x1 + x2 + x3;
        float sq  = x0 * x0 + x1 * x1 + x2 * x2 + x3 * x3;
        for (int off = 16; off > 0; off >>= 1) {
            sum += __shfl_xor(sum, off, 32);
            sq  += __shfl_xor(sq,  off, 32);
        }
        const float mu  = sum * (1.0f / 128.0f);
        const float var = sq * (1.0f / 128.0f) - mu * mu;
        const float rs  = rsqrtf(var + 1e-5f);
        const int row = rowBase + r;
        if (row < N) {
            float* o = out + (long)row * DFEAT;
            o[lane]      = fmaxf((x0 - mu) * rs * gscale[lane]      + gbias[lane],      0.f);
            o[lane + 32] = fmaxf((x1 - mu) * rs * gscale[lane + 32] + gbias[lane + 32], 0.f);
            o[lane + 64] = fmaxf((x2 - mu) * rs * gscale[lane + 64] + gbias[lane + 64], 0.f);
            o[lane + 96] = fmaxf((x3 - mu) * rs * gscale[lane + 96] + gbias[lane + 96], 0.f);
        }
    }
}

extern "C" void kernel_launch(void* const* d_in, const int* in_sizes, int n_in,
                              void* d_out, int out_size, void* d_ws, size_t ws_size,
                              hipStream_t stream) {
    const float* feature  = (const float*)d_in[0];   // [N, 128]
    const float* snorm    = (const float*)d_in[1];   // [N, 1]
    const float* Wm       = (const float*)d_in[2];   // [128, 128]
    const float* ln_scale = (const float*)d_in[3];   // [128]
    const float* ln_bias  = (const float*)d_in[4];   // [128]
    const int*   src      = (const int*)d_in[5];     // [E]
    const int*   dst      = (const int*)d_in[6];     // [E]

    const int N = in_sizes[1];
    const int E = in_sizes[5];

    float* hagg = (float*)d_ws;     // N*128 floats of scratch (25.6 MB)
    float* outp = (float*)d_out;

    // 1) zero the aggregation buffer
    long n4 = ((long)N * DFEAT) >> 2;
    gcn_zero_kernel<<<(int)((n4 + 255) / 256), 256, 0, stream>>>((float4*)hagg, n4);

    // 2) edge-parallel gather + atomic scatter-add (8 edges per 256-thr block)
    gcn_scatter_kernel<<<(E + 7) / 8, 256, 0, stream>>>(feature, src, dst, hagg, E);

    // 3) fused WMMA fp32 GEMM + graph-norm + LayerNorm + ReLU
    gcn_gemm_norm_kernel<<<(N + 15) / 16, 256, 0, stream>>>(
        hagg, Wm, snorm, ln_scale, ln_bias, outp, N);
}